// Qwen3MoeAttention_10187662426954
// MI455X (gfx1250) — compile-verified
//
#include <hip/hip_runtime.h>
#include <hip/hip_bf16.h>
#include <math.h>

// ---------------------------------------------------------------------------
// Qwen3-MoE attention block, fp32, CDNA5 (gfx1250) WMMA implementation.
//   B=2, S=1024, HID=2048, H=32, HK=4, D=128, G=8
// Stages:
//   1) qkv = hidden @ w_qkv          (WMMA f32 16x16x4 GEMM, async-LDS pipe)
//   2) RMSNorm + RoPE on q and k     (VALU, wave-per-head-vector)
//   3) causal GQA flash attention    (WMMA f32 16x16x4, online softmax)
//   4) out = ctx @ w_o               (WMMA f32 16x16x4 GEMM, async-LDS pipe)
// ---------------------------------------------------------------------------

#define CB   2
#define CS   1024
#define CHID 2048
#define CH   32
#define CHK  4
#define CD   128
#define CG   (CH / CHK)
#define CBS  (CB * CS)              // 2048 rows
#define QKV_N ((CH + 2 * CHK) * CD) // 5120
#define CEPS 1e-6f
#define CSCALE 0.08838834764831845f // 1/sqrt(128)

typedef __attribute__((ext_vector_type(2))) float v2f;
typedef __attribute__((ext_vector_type(8))) float v8f;

// --- CDNA5 async global->LDS copy (ASYNCcnt path), with safe fallback ------
#if defined(__AMDGCN__) && __has_builtin(__builtin_amdgcn_global_load_async_to_lds_b128)
#define USE_ASYNC_LDS 1
#else
#define USE_ASYNC_LDS 0
#endif

#if USE_ASYNC_LDS
// Builtin signature (from clang diagnostic): param0 is a non-const GCC-vector
// int4 pointer in AS1 (global); LDS side is the matching AS3 pointer.
typedef int v4i_b __attribute__((vector_size(16)));
typedef __attribute__((address_space(1))) v4i_b* as1_v4i;
typedef __attribute__((address_space(3))) v4i_b* as3_v4i;
#endif

__device__ __forceinline__ void async_copy_b128(void* lds, const void* g) {
#if USE_ASYNC_LDS
  __builtin_amdgcn_global_load_async_to_lds_b128(
      (as1_v4i)(v4i_b*)g, (as3_v4i)(v4i_b*)lds,
      /*offset=*/0, /*cpol=*/0);
#else
  *(float4*)lds = *(const float4*)g;
#endif
}

template <int N>
__device__ __forceinline__ void wait_asynccnt() {
#if USE_ASYNC_LDS
#if __has_builtin(__builtin_amdgcn_s_wait_asynccnt)
  __builtin_amdgcn_s_wait_asynccnt((unsigned short)N);
#else
  asm volatile("s_wait_asynccnt %0" ::"i"(N) : "memory");
#endif
#endif
}

// D = A(16x4) * B(4x16) + C(16x16), all fp32, wave32.
__device__ __forceinline__ v8f wmma4(v2f a, v2f b, v8f c) {
  return __builtin_amdgcn_wmma_f32_16x16x4_f32(
      /*neg_a=*/false, a, /*neg_b=*/false, b,
      /*c_mod=*/(short)0, c, /*reuse_a=*/false, /*reuse_b=*/false);
}

// ---------------------------------------------------------------------------
// fp32 WMMA GEMM: C[M x N] = A[M x K] * B[K x N], row-major.
// Block = 128 threads (4 waves), 64x64 C tile; double-buffered async K-panels.
// Wave w owns rows [w*16, w*16+16) x 64 cols = 4 v8f accumulators.
// ---------------------------------------------------------------------------
__global__ __launch_bounds__(128) void gemm_f32_wmma(
    const float* __restrict__ A, const float* __restrict__ B,
    float* __restrict__ C, int K, int lda, int ldb, int ldc) {
  // Row strides are multiples of 4 floats so every b128 LDS target is 16B-aligned.
  __shared__ __align__(16) float As[2][64][20];  // 64 rows x 16 k (pad 4)
  __shared__ __align__(16) float Bs[2][16][68];  // 16 k x 64 cols (pad 4)

  const int tid  = threadIdx.x;
  const int wave = tid >> 5;
  const int lane = tid & 31;
  const int hl   = lane >> 4;       // lane half (0/1)
  const int ln   = lane & 15;
  const int kk   = hl * 2;          // frag k sub-offset for this half
  const int m0   = blockIdx.x * 64;
  const int n0   = blockIdx.y * 64;

  // Stage one 16-wide K panel into buffer `sel` (4 async b128 per thread).
  auto load_panel = [&](int sel, int k0) {
#pragma unroll
    for (int i = tid; i < 256; i += 128) {   // A: 64x16 tile
      const int row = i >> 2;
      const int c4  = (i & 3) * 4;
      async_copy_b128(&As[sel][row][c4],
                      A + (size_t)(m0 + row) * lda + k0 + c4);
    }
#pragma unroll
    for (int i = tid; i < 256; i += 128) {   // B: 16x64 tile
      const int row = i >> 4;
      const int c4  = (i & 15) * 4;
      async_copy_b128(&Bs[sel][row][c4],
                      B + (size_t)(k0 + row) * ldb + n0 + c4);
    }
  };

  v8f acc[4];
#pragma unroll
  for (int t = 0; t < 4; ++t) acc[t] = (v8f)(0.0f);

  const int nk = K >> 4;
  load_panel(0, 0);  // prologue: panel 0 in flight

  for (int kp = 0; kp < nk; ++kp) {
    const int cur = kp & 1;
    if (kp + 1 < nk) {
      // Issue next panel into the other buffer, then wait until only those
      // 4 newest async ops remain outstanding -> current panel has landed.
      load_panel(cur ^ 1, (kp + 1) << 4);
      wait_asynccnt<4>();
    } else {
      wait_asynccnt<0>();
    }
    __syncthreads();  // all waves' panel writes visible

    const int mrow = wave * 16 + ln;
#pragma unroll
    for (int kc = 0; kc < 4; ++kc) {
      v2f a;
      a[0] = As[cur][mrow][kc * 4 + kk];
      a[1] = As[cur][mrow][kc * 4 + kk + 1];
#pragma unroll
      for (int nt = 0; nt < 4; ++nt) {
        v2f b;
        b[0] = Bs[cur][kc * 4 + kk][nt * 16 + ln];
        b[1] = Bs[cur][kc * 4 + kk + 1][nt * 16 + ln];
        acc[nt] = wmma4(a, b, acc[nt]);
      }
    }
    __syncthreads();  // all reads of buf[cur] done before it is overwritten
  }

  // C layout: VGPR r -> row r (lanes 0-15) / row r+8 (lanes 16-31), col = ln.
#pragma unroll
  for (int nt = 0; nt < 4; ++nt) {
#pragma unroll
    for (int r = 0; r < 8; ++r) {
      const int gm = m0 + wave * 16 + r + hl * 8;
      const int gn = n0 + nt * 16 + ln;
      C[(size_t)gm * ldc + gn] = acc[nt][r];
    }
  }
}

// ---------------------------------------------------------------------------
// Per-head RMSNorm + RoPE, in place. One wave per 128-elem head vector.
// Lane owns elems {2l, 2l+1, 2l+64, 2l+65} so rotate_half is lane-local.
// ---------------------------------------------------------------------------
__global__ __launch_bounds__(128) void rmsnorm_rope(
    float* __restrict__ x, const float* __restrict__ w,
    const float* __restrict__ cosb, const float* __restrict__ sinb,
    int nheads, int row_stride, int nvec) {
  const int wv = blockIdx.x * 4 + (threadIdx.x >> 5);
  if (wv >= nvec) return;
  const int lane = threadIdx.x & 31;
  const int h = wv % nheads;
  const int s = (wv / nheads) % CS;
  const int b = wv / (nheads * CS);

  float* p = x + (size_t)(b * CS + s) * row_stride + h * CD;
  const int i0 = lane * 2;   // 0..62
  const int j0 = i0 + 64;

  float xi0 = p[i0], xi1 = p[i0 + 1], xj0 = p[j0], xj1 = p[j0 + 1];
  float ss = xi0 * xi0 + xi1 * xi1 + xj0 * xj0 + xj1 * xj1;
#pragma unroll
  for (int off = 16; off >= 1; off >>= 1) ss += __shfl_xor(ss, off, 32);
  const float rr = rsqrtf(ss * (1.0f / (float)CD) + CEPS);

  xi0 *= rr * w[i0];     xi1 *= rr * w[i0 + 1];
  xj0 *= rr * w[j0];     xj1 *= rr * w[j0 + 1];

  const float* cs = cosb + (size_t)s * CD;
  const float* sn = sinb + (size_t)s * CD;
  // out = x*cos + rotate_half(x)*sin ; rotate_half = [-x2, x1]
  p[i0]     = xi0 * cs[i0]     - xj0 * sn[i0];
  p[i0 + 1] = xi1 * cs[i0 + 1] - xj1 * sn[i0 + 1];
  p[j0]     = xj0 * cs[j0]     + xi0 * sn[j0];
  p[j0 + 1] = xj1 * cs[j0 + 1] + xi1 * sn[j0 + 1];
}

// ---------------------------------------------------------------------------
// Causal GQA flash attention, one wave (32 threads) per (b, head, 16-row Q tile).
// QK^T and PV both via v_wmma_f32_16x16x4_f32; online softmax in registers.
// Next K/V tile is prefetched (global_prefetch_b8) while current tile computes.
// ---------------------------------------------------------------------------
__global__ __launch_bounds__(32) void flash_attn(
    const float* __restrict__ qkv, float* __restrict__ ctx) {
  __shared__ float pbuf[16][17];  // P tile transpose staging (C-layout -> A-layout)

  const int lane = threadIdx.x;
  const int hl = lane >> 4;
  const int ln = lane & 15;
  const int kk = hl * 2;

  int idx = blockIdx.x;
  const int qt = idx % (CS / 16); idx /= (CS / 16);
  const int h  = idx % CH;        const int b = idx / CH;
  const int kh = h / CG;

  const float* Q  = qkv + (size_t)b * CS * QKV_N + (size_t)h * CD;
  const float* Kp = qkv + (size_t)b * CS * QKV_N + CH * CD + (size_t)kh * CD;
  const float* Vp = qkv + (size_t)b * CS * QKV_N + (CH + CHK) * CD + (size_t)kh * CD;

  // Q tile (16x128) as 32 fp32 A-fragments: lane ln = row, k striped over frag+half.
  v2f qa[32];
  {
    const float* qrow = Q + (size_t)(qt * 16 + ln) * QKV_N;
#pragma unroll
    for (int c = 0; c < 32; ++c) {
      qa[c][0] = qrow[c * 4 + kk];
      qa[c][1] = qrow[c * 4 + kk + 1];
    }
  }

  v8f acc[8];  // ctx 16x128 = 8 N-tiles of 16
#pragma unroll
  for (int t = 0; t < 8; ++t) acc[t] = (v8f)(0.0f);
  float mstat[8], lstat[8];  // row max / row sum; row = r + hl*8
#pragma unroll
  for (int r = 0; r < 8; ++r) { mstat[r] = -3.0e38f; lstat[r] = 0.0f; }

  for (int jt = 0; jt <= qt; ++jt) {
    // ---- prefetch next tile: lanes 0-15 cover K rows, 16-31 cover V rows ----
    if (jt + 1 <= qt) {
      const float* nxt =
          (hl ? Vp : Kp) + (size_t)((jt + 1) * 16 + ln) * QKV_N;
      __builtin_prefetch(nxt, 0, 1);
      __builtin_prefetch(nxt + 32, 0, 1);
      __builtin_prefetch(nxt + 64, 0, 1);
      __builtin_prefetch(nxt + 96, 0, 1);
    }

    // ---- scores = Q @ K^T  (B operand: b[k][n] = K[key=n][d=k]) ----
    v8f sc = (v8f)(0.0f);
    const float* krow = Kp + (size_t)(jt * 16 + ln) * QKV_N;
#pragma unroll
    for (int c = 0; c < 32; ++c) {
      v2f kb;
      kb[0] = krow[c * 4 + kk];
      kb[1] = krow[c * 4 + kk + 1];
      sc = wmma4(qa[c], kb, sc);
    }

    // ---- scale + causal mask (only diagonal tile has partial rows) ----
#pragma unroll
    for (int r = 0; r < 8; ++r) {
      const int M = qt * 16 + r + hl * 8;
      const int N = jt * 16 + ln;
      sc[r] = (N <= M) ? sc[r] * CSCALE : -3.0e38f;
    }

    // ---- online softmax: row reductions within each 16-lane half ----
#pragma unroll
    for (int r = 0; r < 8; ++r) {
      float rm = sc[r];
#pragma unroll
      for (int off = 8; off >= 1; off >>= 1)
        rm = fmaxf(rm, __shfl_xor(rm, off, 32));
      const float mnew  = fmaxf(mstat[r], rm);
      const float alpha = __expf(mstat[r] - mnew);
      const float pv    = __expf(sc[r] - mnew);
      float rs = pv;
#pragma unroll
      for (int off = 8; off >= 1; off >>= 1) rs += __shfl_xor(rs, off, 32);
      lstat[r] = lstat[r] * alpha + rs;
      mstat[r] = mnew;
      sc[r] = pv;
#pragma unroll
      for (int t = 0; t < 8; ++t) acc[t][r] *= alpha;  // rescale ctx row
    }

    // ---- transpose P from C-layout to A-layout through LDS ----
    __syncthreads();
#pragma unroll
    for (int r = 0; r < 8; ++r) pbuf[r + hl * 8][ln] = sc[r];
    __syncthreads();
    v2f pa[4];
#pragma unroll
    for (int c = 0; c < 4; ++c) {
      pa[c][0] = pbuf[ln][c * 4 + kk];
      pa[c][1] = pbuf[ln][c * 4 + kk + 1];
    }

    // ---- ctx += P(16x16) @ V(16x128) ----
#pragma unroll
    for (int c = 0; c < 4; ++c) {
#pragma unroll
      for (int t = 0; t < 8; ++t) {
        v2f vb;
        vb[0] = Vp[(size_t)(jt * 16 + c * 4 + kk) * QKV_N + t * 16 + ln];
        vb[1] = Vp[(size_t)(jt * 16 + c * 4 + kk + 1) * QKV_N + t * 16 + ln];
        acc[t] = wmma4(pa[c], vb, acc[t]);
      }
    }
  }

  // ---- normalize by row sums, store ctx [B,S,H,D] ----
#pragma unroll
  for (int r = 0; r < 8; ++r) {
    const float inv = 1.0f / lstat[r];
    const int gs = qt * 16 + r + hl * 8;
    float* orow = ctx + ((size_t)(b * CS + gs) * (CH * CD)) + h * CD;
#pragma unroll
    for (int t = 0; t < 8; ++t) orow[t * 16 + ln] = acc[t][r] * inv;
  }
}

// ---------------------------------------------------------------------------
extern "C" void kernel_launch(void* const* d_in, const int* in_sizes, int n_in,
                              void* d_out, int out_size, void* d_ws,
                              size_t ws_size, hipStream_t stream) {
  (void)in_sizes; (void)n_in; (void)out_size; (void)ws_size;

  const float* hidden   = (const float*)d_in[0];
  const float* cosb     = (const float*)d_in[1];
  const float* sinb     = (const float*)d_in[2];
  const float* w_qkv    = (const float*)d_in[3];
  const float* q_norm_w = (const float*)d_in[4];
  const float* k_norm_w = (const float*)d_in[5];
  const float* w_o      = (const float*)d_in[6];
  float* out = (float*)d_out;

  float* ws  = (float*)d_ws;
  float* qkv = ws;                            // CBS * QKV_N  (40 MB)
  float* ctx = ws + (size_t)CBS * QKV_N;      // CBS * H*D    (32 MB)

  // 1) QKV projection: [2048 x 2048] @ [2048 x 5120]
  gemm_f32_wmma<<<dim3(CBS / 64, QKV_N / 64), dim3(128), 0, stream>>>(
      hidden, w_qkv, qkv, CHID, CHID, QKV_N, QKV_N);

  // 2) RMSNorm + RoPE on q (32 heads) and k (4 heads), in place in qkv.
  rmsnorm_rope<<<dim3((CB * CS * CH) / 4), dim3(128), 0, stream>>>(
      qkv, q_norm_w, cosb, sinb, CH, QKV_N, CB * CS * CH);
  rmsnorm_rope<<<dim3((CB * CS * CHK) / 4), dim3(128), 0, stream>>>(
      qkv + CH * CD, k_norm_w, cosb, sinb, CHK, QKV_N, CB * CS * CHK);

  // 3) causal GQA flash attention -> ctx [B,S,H,D]
  flash_attn<<<dim3(CB * CH * (CS / 16)), dim3(32), 0, stream>>>(qkv, ctx);

  // 4) output projection: [2048 x 4096] @ [4096 x 2048]
  gemm_f32_wmma<<<dim3(CBS / 64, CHID / 64), dim3(128), 0, stream>>>(
      ctx, w_o, out, CH * CD, CH * CD, CHID, CHID);
}